// SpaNn_57853209477376
// MI455X (gfx1250) — compile-verified
//
#include <hip/hip_runtime.h>
#include <math.h>
#include <stdint.h>

// SPA-NN belief-propagation decoder, sparse formulation.
// N=1024 vars (deg 4), M=512 checks (deg 8), E=4096 edges.
#define N_VARS  1024
#define DV      4
#define DC      8
#define E_EDGES 4096
#define EPS32   1.1920929e-07f

typedef unsigned int u32x4 __attribute__((ext_vector_type(4)));
typedef int          i32x4 __attribute__((ext_vector_type(4)));
typedef int          i32x8 __attribute__((ext_vector_type(8)));

// --- fast transcendentals: CDNA5 has hardware V_TANH_F32 (TRANS unit) ------
#if __has_builtin(__builtin_amdgcn_tanhf)
#define FAST_TANH(x) __builtin_amdgcn_tanhf(x)
#else
#define FAST_TANH(x) tanhf(x)
#endif

#if __has_builtin(__builtin_amdgcn_rcpf)
#define FAST_RCP(x) __builtin_amdgcn_rcpf(x)
#else
#define FAST_RCP(x) (1.0f / (x))
#endif

// ---------------------------------------------------------------------------
// Phase 1: extract, for every edge row i of w_odd_to_even [E x E], the 7
// partner edges j with w[i,j] != 0 (edges sharing i's check node).
// One wave32 per row; coalesced float4 loads; ballot+popcount compaction.
// Output: part[row*DC + 0..6] (int), stride DC=8.
// ---------------------------------------------------------------------------
__global__ void spa_extract_partners(const float* __restrict__ w,
                                     int* __restrict__ part) {
    const int wavesPerBlock = blockDim.x >> 5;
    const int row  = blockIdx.x * wavesPerBlock + ((int)threadIdx.x >> 5);
    const int lane = (int)threadIdx.x & 31;
    if (row >= E_EDGES) return;

    const float4* r4 = (const float4*)(w + (size_t)row * E_EDGES);
    int count = 0;
    for (int c = 0; c < E_EDGES / 128; ++c) {          // 32 chunks of 128 elems
        const float4 v = r4[c * 32 + lane];
        const float vals[4] = {v.x, v.y, v.z, v.w};
#pragma unroll
        for (int k = 0; k < 4; ++k) {
            const bool hit = (vals[k] != 0.0f);
            const unsigned m = __builtin_amdgcn_ballot_w32(hit);
            if (hit) {
                const int pos =
                    count + __builtin_popcount(m & ((1u << lane) - 1u));
                if (pos < DC)
                    part[row * DC + pos] = (c * 32 + lane) * 4 + k;
            }
            count += __builtin_popcount(m);            // wave-uniform
        }
    }
}

// ---------------------------------------------------------------------------
// Phase 2: single workgroup (1024 threads = 32 waves on one WGP) runs the
// whole 5-iteration decode in LDS. Thread t owns variable t and its 4 edges
// e = 4t..4t+3 (var_of_edge[e] = e>>2 analytically), so the variable-node
// update is register-local; only the check-node product gathers from LDS.
// x is DMA'd into LDS with the Tensor Data Mover.
// LDS layout (dynamic): lx[1024] f32 @0 (4KB) | odd[4096] f32 (16KB).
// ---------------------------------------------------------------------------
__global__ void __launch_bounds__(N_VARS) spa_iterate(
    const float* __restrict__ x,
    const int*   __restrict__ part,
    float*       __restrict__ out) {

    extern __shared__ unsigned char smem[];
    float* lx  = (float*)smem;                 // LDS byte offset 0
    float* odd = (float*)(smem + 4096);

    const int t = (int)threadIdx.x;

    // --- TDM: async DMA of x (1024 f32 = 4KB) into LDS offset 0 -----------
    if (t == 0) {
        const unsigned long long ga = (unsigned long long)(uintptr_t)x;
        // D# group0: count=1 | lds_addr=0 | global_addr | type=2
        const u32x4 g0 = { 1u,
                           0u,
                           (unsigned)(ga & 0xFFFFFFFFu),
                           ((unsigned)(ga >> 32) & 0x01FFFFFFu) | 0x80000000u };
        // D# group1: data_size=4B(code 2); tensor_dim0=1024; tensor_dim1=1;
        //            tile_dim0=1024; tile_dim1=1; tensor_dim0_stride=1024.
        const i32x8 g1 = { (int)0x00020000,        // mask=0, data_size=2
                           (int)(1024u << 16),     // tensor_dim0 lo16 @bit48
                           (int)(1u    << 16),     // tensor_dim1 lo16 @bit80
                           (int)(1024u << 16),     // tile_dim0      @bit112
                           1,                      // tile_dim1=1, tile_dim2=0
                           1024,                   // tensor_dim0_stride lo32
                           0, 0 };
        const i32x4 gz = { 0, 0, 0, 0 };
#if defined(__clang_major__) && (__clang_major__ >= 23)
        const i32x8 gz8 = { 0, 0, 0, 0, 0, 0, 0, 0 };
        __builtin_amdgcn_tensor_load_to_lds(g0, g1, gz, gz, gz8, 0);
#else
        __builtin_amdgcn_tensor_load_to_lds(g0, g1, gz, gz, 0);
#endif
        __builtin_amdgcn_s_wait_tensorcnt(0);
    }

    // --- partner indices for my 4 edges -> registers (L2-resident, 128KB) --
    unsigned short pl[DV][DC - 1];
#pragma unroll
    for (int k = 0; k < DV; ++k) {
        const int e = t * DV + k;
#pragma unroll
        for (int j = 0; j < DC - 1; ++j)
            pl[k][j] = (unsigned short)part[e * DC + j];
    }
    __syncthreads();                         // TDM data visible to all waves

    const float myx = lx[t];
    float ev[DV];

    // layer-1 odd message: even input is zero
    {
        const float o = FAST_TANH(0.5f * myx);
#pragma unroll
        for (int k = 0; k < DV; ++k) odd[t * DV + k] = o;
    }

    for (int it = 0; it < 5; ++it) {
        __syncthreads();                     // odd stores visible to all
        // check-node (even) update: extrinsic product of 7 partner odds,
        // then 2*atanh(p) = ln((1+p)/(1-p))  (v_log_f32 + fast divide)
#pragma unroll
        for (int k = 0; k < DV; ++k) {
            float p = 1.0f;
#pragma unroll
            for (int j = 0; j < DC - 1; ++j)
                p *= odd[pl[k][j]];
            p = (p >=  1.0f) ? ( 1.0f - EPS32) : p;
            p = (p <= -1.0f) ? (-1.0f + EPS32) : p;
            ev[k] = __logf((1.0f + p) * FAST_RCP(1.0f - p));
        }
        __syncthreads();                     // gathers done before overwrite
        if (it < 4) {
            // variable-node (odd) update: extrinsic sum = S - ev[k], all local
            const float S = ev[0] + ev[1] + ev[2] + ev[3];
#pragma unroll
            for (int k = 0; k < DV; ++k)
                odd[t * DV + k] = FAST_TANH(0.5f * (myx + S - ev[k]));
        }
    }

    // marginal: out[t] = sigmoid(x[t] + sum of final even messages of var t)
    const float z = myx + ev[0] + ev[1] + ev[2] + ev[3];
    out[t] = FAST_RCP(1.0f + __expf(-z));
}

// ---------------------------------------------------------------------------
extern "C" void kernel_launch(void* const* d_in, const int* in_sizes, int n_in,
                              void* d_out, int out_size, void* d_ws,
                              size_t ws_size, hipStream_t stream) {
    (void)in_sizes; (void)n_in; (void)out_size; (void)ws_size;

    const float* x      = (const float*)d_in[0];  // [1024]
    const float* w_o2e  = (const float*)d_in[2];  // [4096*4096] check mask
    float*       out    = (float*)d_out;          // [1024]
    int*         part   = (int*)d_ws;             // [4096*8] partner indices

    // Phase 1: 4096 rows, one wave32 each -> 8 waves/block, 512 blocks.
    spa_extract_partners<<<dim3(E_EDGES / 8), dim3(256), 0, stream>>>(w_o2e,
                                                                      part);

    // Phase 2: one WGP-resident workgroup; 20KB dynamic LDS.
    const size_t smem = 4096 /*lx*/ + (size_t)E_EDGES * 4 /*odd*/;
    spa_iterate<<<dim3(1), dim3(N_VARS), smem, stream>>>(x, part, out);
}